// TransformerEncoderHFPrompt_64879775973533
// MI455X (gfx1250) — compile-verified
//
#include <hip/hip_runtime.h>

// Problem constants (from reference setup_inputs)
#define BB 8
#define SS 4096
#define HH 768
#define DD 1024
#define WW 2048  // MAX_WORDS

typedef __attribute__((ext_vector_type(2))) float v2f;
typedef __attribute__((ext_vector_type(4))) float v4f;
typedef __attribute__((ext_vector_type(8))) float v8f;

// ---------------------------------------------------------------------------
// Kernel 1: segment-mean pooling (deterministic: binary search on sorted ids,
// no float atomics). One block per (batch, word). Writes pooled token
// embeddings (B*W, H) to workspace and token_masks (int32) to d_out tail.
// ---------------------------------------------------------------------------
__global__ __launch_bounds__(256) void word_pool_kernel(
    const float* __restrict__ emb,      // (B, S, H)
    const int*   __restrict__ masks,    // (B, S)
    const int*   __restrict__ word_ids, // (B, S) sorted along S
    float*       __restrict__ tokemb,   // (B*W, H) workspace
    int*         __restrict__ token_masks) // (B*W)
{
    const int blk = blockIdx.x;
    const int b = blk >> 11;      // / WW
    const int w = blk & (WW - 1); // % WW

    const int* wid = word_ids + (size_t)b * SS;

    // lower_bound(w)
    int lo = 0, hi = SS;
    while (lo < hi) { int mid = (lo + hi) >> 1; if (wid[mid] <  w) lo = mid + 1; else hi = mid; }
    const int s0 = lo;
    // upper_bound(w)
    hi = SS;
    while (lo < hi) { int mid = (lo + hi) >> 1; if (wid[mid] <= w) lo = mid + 1; else hi = mid; }
    const int s1 = lo;

    const int count = s1 - s0;
    int msum = 0;
    const int* mrow = masks + (size_t)b * SS;
    for (int s = s0; s < s1; ++s) msum += mrow[s];

    const bool valid = (count > 0) && (msum == count);
    const float inv = valid ? (1.0f / (float)count) : 0.0f; // invalid -> zeros

    float* dst = tokemb + ((size_t)(b * WW + w)) * HH;
    const float* src = emb + ((size_t)b * SS + (size_t)s0) * HH;

    for (int d = threadIdx.x; d < HH; d += 256) {
        float acc = 0.0f;
        for (int s = 0; s < count; ++s) acc += src[(size_t)s * HH + d];
        dst[d] = acc * inv;
    }
    if (threadIdx.x == 0) token_masks[b * WW + w] = valid ? 1 : 0;
}

// ---------------------------------------------------------------------------
// Kernel 2: fp32 WMMA GEMM  out(M=16384, N=1024) = T(M,768) * Wp(768,1024) + b
// Block: 256 threads = 8 waves. Block tile 128x64, wave tile 32x32.
// K staged through LDS in chunks of 16; V_WMMA_F32_16X16X4_F32 per 16x16x4.
//
// LDS layouts (64 banks x 4B):
//   A: row-major, stride LDA=20 floats -> per-lane row-strided frag reads hit
//      16 distinct banks; the two K values per lane are one aligned b64.
//   B: K-pair interleaved: (k,n) -> sB[(k>>1)*LDBP + 2n + (k&1)], LDBP=160
//      (mod 64 == 32). A lane's fragment {B[2kp][n], B[2kp+1][n]} is one
//      aligned b64; lane halves (kp vs kp+1) land in disjoint 32-bank halves.
// ---------------------------------------------------------------------------
#define MBLK 128
#define NBLK 64
#define KBLK 16
#define LDA  20    // floats
#define LDBP 160   // floats per K-pair row (64 pairs*2 + 32 pad)

__global__ __launch_bounds__(256) void proj_gemm_wmma_f32(
    const float* __restrict__ A,    // (16384, 768) pooled tokens, row-major
    const float* __restrict__ Wp,   // (768, 1024) row-major
    const float* __restrict__ bias, // (1024)
    float*       __restrict__ out)  // (16384, 1024)
{
    __shared__ float sA[MBLK * LDA];            // 10240 B
    __shared__ float sB[(KBLK / 2) * LDBP];     //  5120 B

    const int tid  = threadIdx.x;
    const int M0   = blockIdx.y * MBLK;
    const int N0   = blockIdx.x * NBLK;
    const int wid  = tid >> 5;
    const int lane = tid & 31;
    const int h    = lane >> 4;   // lane half: K-pair select per ISA frag layout
    const int l16  = lane & 15;
    const int mW   = (wid & 3) * 32;  // 4 waves along M
    const int nW   = (wid >> 2) * 32; // 2 waves along N

    v8f acc[2][2] = {};

    // cooperative loader indices
    const int arow = tid >> 1;         // A: 128 rows, 2 threads/row
    const int acol = (tid & 1) * 8;    //    8 floats each (2x float4)
    const int bkp  = tid >> 5;         // B: k-pair 0..7
    const int bn2  = (tid & 31) * 2;   //    even n within 64-wide tile

    for (int k0 = 0; k0 < HH; k0 += KBLK) {
        // stage A tile (128 x 16), row-major
        const float* ga = A + (size_t)(M0 + arow) * HH + k0 + acol;
        v4f a0 = *(const v4f*)(ga);
        v4f a1 = *(const v4f*)(ga + 4);
        *(v4f*)(&sA[arow * LDA + acol])     = a0;
        *(v4f*)(&sA[arow * LDA + acol + 4]) = a1;

        // stage B tile (16 x 64), K-pair interleaved
        const float* gb = Wp + (size_t)(k0 + 2 * bkp) * DD + N0 + bn2;
        v2f r0 = *(const v2f*)(gb);        // row 2kp,   cols n2..n2+1
        v2f r1 = *(const v2f*)(gb + DD);   // row 2kp+1, cols n2..n2+1
        v4f bq; bq[0] = r0[0]; bq[1] = r1[0]; bq[2] = r0[1]; bq[3] = r1[1];
        *(v4f*)(&sB[bkp * LDBP + bn2 * 2]) = bq;

        // prefetch next chunk's global lines (global_prefetch_b8)
        if (k0 + KBLK < HH) {
            __builtin_prefetch(ga + KBLK, 0, 3);
            __builtin_prefetch(gb + (size_t)KBLK * DD, 0, 3);
        }
        __syncthreads();

#pragma unroll
        for (int kk = 0; kk < KBLK; kk += 4) {
            v2f af[2], bf[2];
            // A fragment: 16x4 f32 layout — lane l16 = row M; lane half h picks
            // K=2h+j in VGPR j  (ISA 7.12.2, 32-bit A matrix) — aligned b64
#pragma unroll
            for (int mt = 0; mt < 2; ++mt)
                af[mt] = *(const v2f*)(&sA[(mW + mt * 16 + l16) * LDA + kk + 2 * h]);
            // B fragment: one aligned b64 from the interleaved layout
#pragma unroll
            for (int nt = 0; nt < 2; ++nt)
                bf[nt] = *(const v2f*)(&sB[((kk >> 1) + h) * LDBP
                                           + (nW + nt * 16 + l16) * 2]);
#pragma unroll
            for (int mt = 0; mt < 2; ++mt)
#pragma unroll
                for (int nt = 0; nt < 2; ++nt)
                    acc[mt][nt] = __builtin_amdgcn_wmma_f32_16x16x4_f32(
                        /*neg_a=*/false, af[mt], /*neg_b=*/false, bf[nt],
                        /*c_mod=*/(short)0, acc[mt][nt],
                        /*reuse_a=*/false, /*reuse_b=*/false);
        }
        __syncthreads();
    }

    // epilogue: C/D 16x16 f32 layout — VGPR r holds M=r (lanes 0-15) / M=r+8
    // (lanes 16-31), N = l16. Add bias, store.
#pragma unroll
    for (int nt = 0; nt < 2; ++nt) {
        const int col = N0 + nW + nt * 16 + l16;
        const float bv = bias[col];
#pragma unroll
        for (int mt = 0; mt < 2; ++mt) {
#pragma unroll
            for (int r = 0; r < 8; ++r) {
                const int row = M0 + mW + mt * 16 + r + 8 * h;
                out[(size_t)row * DD + col] = acc[mt][nt][r] + bv;
            }
        }
    }
}

// ---------------------------------------------------------------------------
// Launch: inputs in setup_inputs() order:
//   0: embeddings (B,S,H) f32   1: proj_w (H,D) f32   2: proj_b (D,) f32
//   3: masks (B,S) i32          4: word_ids (B,S) i32
// Output: [ out (B,W,D) f32 | token_masks (B,W) i32 ] concatenated in d_out.
// ---------------------------------------------------------------------------
extern "C" void kernel_launch(void* const* d_in, const int* in_sizes, int n_in,
                              void* d_out, int out_size, void* d_ws, size_t ws_size,
                              hipStream_t stream) {
    const float* emb      = (const float*)d_in[0];
    const float* proj_w   = (const float*)d_in[1];
    const float* proj_b   = (const float*)d_in[2];
    const int*   masks    = (const int*)d_in[3];
    const int*   word_ids = (const int*)d_in[4];

    float* out = (float*)d_out;
    int* token_masks = (int*)(out + (size_t)BB * WW * DD);
    float* tokemb = (float*)d_ws;  // (B*W, H) = 50.3 MB scratch

    word_pool_kernel<<<BB * WW, 256, 0, stream>>>(emb, masks, word_ids,
                                                  tokemb, token_masks);

    dim3 grid(DD / NBLK, (BB * WW) / MBLK);  // (16, 128)
    proj_gemm_wmma_f32<<<grid, 256, 0, stream>>>(tokemb, proj_w, proj_b, out);
}